// AS_RNN_20134806683794
// MI455X (gfx1250) — compile-verified
//
#include <hip/hip_runtime.h>
#include <cmath>

// ---------------------------------------------------------------------------
// Leaky RNN scan:  U = Lam*(R@W^T + b) + (1-Lam)*[x_t, 0];  R <- tanh(U)
// T=8192 serial steps, W is 4096x4096 fp32 (64 MB -> resident in MI455X 192MB L2).
// Strategy: persistent kernel, 256 wave32 waves, each owning one 16-row M-tile.
// Matvec via V_WMMA_F32_16X16X4_F32 (A = W tile, B = R broadcast to 16 cols).
// One device-wide atomic barrier per step; R double-buffered in d_ws.
// ---------------------------------------------------------------------------

constexpr int   VISIBLE = 1024;
constexpr int   FULL    = 4096;   // 1024 visible + 3072 hidden
constexpr int   T_STEPS = 8192;
constexpr float LAM     = 0.5f;

constexpr int NBLOCKS = 64;
constexpr int WPB     = 4;              // waves per block
constexpr int THREADS = WPB * 32;       // wave32
constexpr int NWAVES  = NBLOCKS * WPB;  // 256
static_assert(FULL / 16 == NWAVES, "exactly one 16-row M-tile per wave");

typedef float v2f __attribute__((ext_vector_type(2)));
typedef float v8f __attribute__((ext_vector_type(8)));

__global__ __launch_bounds__(THREADS, 1)
void AS_RNN_persistent(const float* __restrict__ X,   // [T, VISIBLE]
                       const float* __restrict__ W,   // [FULL, FULL] row-major
                       const float* __restrict__ b,   // [FULL]
                       float*       __restrict__ out, // [VISIBLE]
                       float*       __restrict__ Rbuf,// [2, FULL] in d_ws (zeroed)
                       unsigned*    __restrict__ cnt) // barrier counter (zeroed)
{
  __shared__ float sR[FULL];                          // 16 KB of 320 KB LDS

  const int tid  = threadIdx.x;
  const int lane = tid & 31;
  const int half = lane >> 4;          // 0: lanes 0-15, 1: lanes 16-31
  const int koff = half * 2;           // K offset inside a 16x16x4 fragment
  const int wave = tid >> 5;
  const int m0   = (blockIdx.x * WPB + wave) * 16;    // this wave's 16 rows

  // A-fragment base: lane l streams row m0+(l&15), columns koff, koff+1, ...
  const float* __restrict__ wrow =
      W + (size_t)(m0 + (lane & 15)) * FULL + koff;

  for (int step = 0; step < T_STEPS; ++step) {
    const float* __restrict__ Rin  = Rbuf + ((step    ) & 1) * FULL;
    float*       __restrict__ Rout = Rbuf + ((step + 1) & 1) * FULL;

    // Stage R (16 KB) into LDS cooperatively (b128 loads / ds b128 stores).
    for (int i = tid; i < FULL / 4; i += THREADS)
      reinterpret_cast<float4*>(sR)[i] =
          reinterpret_cast<const float4*>(Rin)[i];
    __syncthreads();

    // K reduction: 1024 x (16x16x4) fp32 WMMAs, two accumulator chains.
    v8f c0 = {}; v8f c1 = {};
    const float* wp = wrow;
#pragma unroll 4
    for (int k = 0; k < FULL; k += 8) {
      v2f a0 = *reinterpret_cast<const v2f*>(wp);              // global b64
      v2f b0 = *reinterpret_cast<const v2f*>(&sR[k + koff]);   // ds b64
      c0 = __builtin_amdgcn_wmma_f32_16x16x4_f32(false, a0, false, b0,
                                                 (short)0, c0, false, false);
      v2f a1 = *reinterpret_cast<const v2f*>(wp + 4);
      v2f b1 = *reinterpret_cast<const v2f*>(&sR[k + 4 + koff]);
      c1 = __builtin_amdgcn_wmma_f32_16x16x4_f32(false, a1, false, b1,
                                                 (short)0, c1, false, false);
      wp += 8;
    }
    const v8f c = c0 + c1;

    // B was broadcast -> every D column identical. Lane 0 holds rows m0..m0+7
    // in c[0..7]; lane 16 holds rows m0+8..m0+15. Those two lanes finalize.
    if ((lane & 15) == 0) {
      const int base = m0 + half * 8;
#pragma unroll
      for (int r = 0; r < 8; ++r) {
        const int i = base + r;
        float u = c[r] + b[i];
        if (i < VISIBLE)
          u = LAM * u + (1.0f - LAM) * X[(size_t)step * VISIBLE + i];
        if (i < VISIBLE && step == T_STEPS - 1)
          out[i] = u;                       // reference returns pre-tanh U
        Rout[i] = tanhf(u);
      }
    }

    // Device-wide step barrier (cumulative counter; no reset needed).
    __builtin_amdgcn_fence(__ATOMIC_RELEASE, "agent");  // flush Rout stores
    __syncthreads();
    if (tid == 0) {
      atomicAdd(cnt, 1u);
      const unsigned target = (unsigned)(step + 1) * (unsigned)NBLOCKS;
      while (__hip_atomic_load(cnt, __ATOMIC_RELAXED,
                               __HIP_MEMORY_SCOPE_AGENT) < target)
        __builtin_amdgcn_s_sleep(1);
    }
    __syncthreads();
    __builtin_amdgcn_fence(__ATOMIC_ACQUIRE, "agent");  // invalidate L0s
  }
}

extern "C" void kernel_launch(void* const* d_in, const int* in_sizes, int n_in,
                              void* d_out, int out_size, void* d_ws, size_t ws_size,
                              hipStream_t stream) {
  const float* X = (const float*)d_in[0];   // [8192, 1024] fp32
  const float* W = (const float*)d_in[1];   // [4096, 4096] fp32
  const float* b = (const float*)d_in[2];   // [4096] fp32

  float*    Rbuf = (float*)d_ws;                                   // 2*FULL f32
  unsigned* cnt  = (unsigned*)((char*)d_ws + 2 * FULL * sizeof(float));

  // Zero R double-buffer (R0 = 0) and barrier counter each launch —
  // graph-capture-safe (memset node), deterministic across replays.
  hipMemsetAsync(d_ws, 0, 2 * FULL * sizeof(float) + 256, stream);

  AS_RNN_persistent<<<NBLOCKS, THREADS, 0, stream>>>(
      X, W, b, (float*)d_out, Rbuf, cnt);
}